// ImaginaryDimensionAdder_8340826489317
// MI455X (gfx1250) — compile-verified
//
#include <hip/hip_runtime.h>

// ImaginaryDimensionAdder: out[:, 2i] = in[:, i]; out[:, 2i+1] = 0
// Pure streaming op: 128 MiB read + 256 MiB write @ 23.3 TB/s => ~17us floor.
// Per lane: load float2 (global_load_b64 th:NT), store float4 {r.x,0,r.y,0}
// (global_store_b128 th:NT). Fast/tail split is UNIFORM per block (depends
// only on blockIdx) so it lowers to a scalar s_cbranch; the fast path is
// straight-line: 8 loads in flight on LOADcnt, staggered waits, 8 stores.

typedef __attribute__((ext_vector_type(2))) float v2f;
typedef __attribute__((ext_vector_type(4))) float v4f;

#define THREADS 256u
#define UNROLL  8u
#define PER_BLOCK (THREADS * UNROLL)

__global__ __launch_bounds__(THREADS) void
ImaginaryDimensionAdder_8340826489317_kernel(const v2f* __restrict__ in,
                                             v4f* __restrict__ out,
                                             unsigned int n2 /* # of float2 elems */) {
    unsigned int blockBase = blockIdx.x * PER_BLOCK;

    if (blockBase + PER_BLOCK <= n2) {
        // Fast path — uniform (scalar) branch condition, no EXEC masking.
        // Whole tile in range: issue all loads, then all stores.
        unsigned int base = blockBase + threadIdx.x;
        v2f r[UNROLL];
#pragma unroll
        for (unsigned int k = 0; k < UNROLL; ++k) {
            r[k] = __builtin_nontemporal_load(&in[base + k * THREADS]);
        }
#pragma unroll
        for (unsigned int k = 0; k < UNROLL; ++k) {
            v4f o;
            o.x = r[k].x;
            o.y = 0.0f;
            o.z = r[k].y;
            o.w = 0.0f;
            __builtin_nontemporal_store(o, &out[base + k * THREADS]);
        }
    } else {
        // Tail path — only the final block can land here; never taken for the
        // 8192x4096 shape (16,777,216 float2 divides evenly by 2048).
        unsigned int base = blockBase + threadIdx.x;
        for (unsigned int k = 0; k < UNROLL; ++k) {
            unsigned int i = base + k * THREADS;
            if (i < n2) {
                v2f r = __builtin_nontemporal_load(&in[i]);
                v4f o;
                o.x = r.x;
                o.y = 0.0f;
                o.z = r.y;
                o.w = 0.0f;
                __builtin_nontemporal_store(o, &out[i]);
            }
        }
    }
}

extern "C" void kernel_launch(void* const* d_in, const int* in_sizes, int n_in,
                              void* d_out, int out_size, void* d_ws, size_t ws_size,
                              hipStream_t stream) {
    (void)n_in; (void)out_size; (void)d_ws; (void)ws_size;

    const v2f* in = (const v2f*)d_in[0];
    v4f* out = (v4f*)d_out;

    // in_sizes[0] = total float elements (8192*4096). Each work item is one
    // float2 of input producing one float4 of output.
    unsigned int n_floats = (unsigned int)in_sizes[0];
    unsigned int n2 = n_floats / 2u;

    unsigned int blocks = (n2 + PER_BLOCK - 1u) / PER_BLOCK;

    ImaginaryDimensionAdder_8340826489317_kernel<<<blocks, THREADS, 0, stream>>>(in, out, n2);
}